// QuantumAttentionBlock_65481071401073
// MI455X (gfx1250) — compile-verified
//
#include <hip/hip_runtime.h>
#include <hip/hip_bf16.h>

// ---------------------------------------------------------------------------
// Quantum circuit sim restructured as:
//   stage 1: build fixed 256x256 complex unitary U (shared gates) -> d_ws
//            stored TRANSPOSED: Ut[j][n] = U[n][j]  (so GEMM B-loads coalesce)
//   stage 2: per sample, closed-form product init state (8 RX on |0>), then
//            complex GEMM  out[m][n] = sum_j init[m][j] * Ut[j][n]
//            via V_WMMA_F32_16X16X4_F32, then |.|^2 and +-1 sign reduction.
// ---------------------------------------------------------------------------

typedef float v2f __attribute__((ext_vector_type(2)));
typedef float v8f __attribute__((ext_vector_type(8)));

#define N_WIRES 8
#define DIM 256
#define N_ROPS 20
#define COLS_PER_BLOCK 16   // stage-1: identity columns per block

// --- GATE_OPS generated from np.random.default_rng(42) (see reference):
//     kind: 0=RX 1=RY 2=RZ 3=CNOT(a=control, b=target). Regenerate host-side
//     if the seed/ordering changes; structure is compile-time constant.
__device__ __constant__ int G_KIND[N_ROPS] = {0,3,2,1,3,1,0,3,2,0,3,1,2,3,0,1,3,2,0,1};
__device__ __constant__ int G_A   [N_ROPS] = {6,4,2,0,1,3,7,2,5,1,6,4,0,5,3,7,0,4,5,2};
__device__ __constant__ int G_B   [N_ROPS] = {-1,7,-1,-1,5,-1,-1,0,-1,-1,3,-1,-1,2,-1,-1,6,-1,-1,-1};

// ============================================================================
// Stage 1: apply the 20 shared ops + 8 final RX to identity columns.
// Block b owns columns j = b*16 .. b*16+15; state kept in LDS (32 KB).
// Writes Ut row j (= column j of U) contiguously -> coalesced.
// ============================================================================
__global__ __launch_bounds__(256) void build_unitary(
    const float* __restrict__ rx_params,
    const float* __restrict__ random_params,
    float* __restrict__ ut_re, float* __restrict__ ut_im) {
  __shared__ float sr[COLS_PER_BLOCK][DIM];
  __shared__ float si[COLS_PER_BLOCK][DIM];
  const int tid = threadIdx.x;
  const int jbase = blockIdx.x * COLS_PER_BLOCK;

  for (int idx = tid; idx < COLS_PER_BLOCK * DIM; idx += 256) {
    int c = idx >> 8, r = idx & 255;
    sr[c][r] = (r == jbase + c) ? 1.f : 0.f;
    si[c][r] = 0.f;
  }
  __syncthreads();

  for (int p = 0; p < N_ROPS + N_WIRES; ++p) {
    int kind, a, b; float theta;
    if (p < N_ROPS) { kind = G_KIND[p]; a = G_A[p]; b = G_B[p]; theta = random_params[p]; }
    else            { kind = 0; a = p - N_ROPS; b = -1; theta = rx_params[p - N_ROPS]; }

    if (kind == 3) {                              // CNOT(control=a, target=b)
      const int cpos = 7 - a, tpos = 7 - b;
      const int cmask = 1 << cpos, tmask = 1 << tpos;
      const int plo = (cpos < tpos) ? cpos : tpos;
      const int phi = (cpos < tpos) ? tpos : cpos;
      for (int q = tid; q < COLS_PER_BLOCK * 64; q += 256) {
        int col = q >> 6, pr = q & 63;
        int r = pr;
        r = ((r & ~((1 << plo) - 1)) << 1) | (r & ((1 << plo) - 1));
        r = ((r & ~((1 << phi) - 1)) << 1) | (r & ((1 << phi) - 1));
        r |= cmask;                               // control=1, target=0
        int r1 = r | tmask;
        float tr = sr[col][r], ti = si[col][r];
        sr[col][r] = sr[col][r1]; si[col][r] = si[col][r1];
        sr[col][r1] = tr;         si[col][r1] = ti;
      }
    } else {
      const float h = theta * 0.5f;
      const float c = cosf(h), s = sinf(h);
      float g00r, g00i, g01r, g01i, g10r, g10i, g11r, g11i;
      if (kind == 0)      { g00r=c; g00i=0; g01r=0;  g01i=-s; g10r=0; g10i=-s; g11r=c; g11i=0; }  // RX
      else if (kind == 1) { g00r=c; g00i=0; g01r=-s; g01i=0;  g10r=s; g10i=0;  g11r=c; g11i=0; }  // RY
      else                { g00r=c; g00i=-s; g01r=0; g01i=0;  g10r=0; g10i=0;  g11r=c; g11i=s; }  // RZ
      const int mask = 1 << (7 - a);
      for (int q = tid; q < COLS_PER_BLOCK * 128; q += 256) {
        int col = q >> 7, pr = q & 127;
        int r0 = ((pr & ~(mask - 1)) << 1) | (pr & (mask - 1));
        int r1 = r0 | mask;
        float ar = sr[col][r0], ai = si[col][r0];
        float br = sr[col][r1], bi = si[col][r1];
        sr[col][r0] = g00r*ar - g00i*ai + g01r*br - g01i*bi;
        si[col][r0] = g00r*ai + g00i*ar + g01r*bi + g01i*br;
        sr[col][r1] = g10r*ar - g10i*ai + g11r*br - g11i*bi;
        si[col][r1] = g10r*ai + g10i*ar + g11r*bi + g11i*br;
      }
    }
    __syncthreads();
  }

  for (int idx = tid; idx < COLS_PER_BLOCK * DIM; idx += 256) {
    int c = idx >> 8, n = idx & 255;
    ut_re[(long)(jbase + c) * DIM + n] = sr[c][n];
    ut_im[(long)(jbase + c) * DIM + n] = si[c][n];
  }
}

// ============================================================================
// Stage 2: 16 samples / block, 256 threads (8 wave32).
//   LDS A:  init state transposed [j][m]  (conflict-free ds loads for A frags)
//   GEMM:   each wave owns N-tiles {wid, wid+8}; K-loop step 4;
//           4 accumulators rr/ii/ri/ir (f32 WMMA has no A/B negate).
//   Epilogue: probs -> LDS, then 128 threads do the +-1 sign reduction.
// ============================================================================
#define PROBW 260  // padded row stride for probs

__global__ __launch_bounds__(256) void qsim_wmma(
    const float* __restrict__ x,
    const float* __restrict__ ut_re, const float* __restrict__ ut_im,
    float* __restrict__ out) {
  __shared__ float sRe[DIM * 16];        // [j][m]
  __shared__ float sIm[DIM * 16];
  __shared__ float sProb[16 * PROBW];    // [m][n]

  const int tid = threadIdx.x;
  const long base = (long)blockIdx.x * 16;

  // ---- closed-form product init state -------------------------------------
  {
    const int m = tid >> 4;      // sample within block
    const int g = tid & 15;      // j-group (16 amplitudes per thread)
    float cw[8], sw[8];
    const float* xs = x + (base + m) * 8;
#pragma unroll
    for (int w = 0; w < 8; ++w) {
      float h = xs[w] * 0.5f;
      cw[w] = __cosf(h);
      sw[w] = __sinf(h);
    }
#pragma unroll
    for (int jj = 0; jj < 16; ++jj) {
      int j = g * 16 + jj;
      float p = 1.f;
#pragma unroll
      for (int w = 0; w < 8; ++w)
        p *= ((j >> (7 - w)) & 1) ? sw[w] : cw[w];
      int n1 = __popc(j) & 3;                 // phase (-i)^popcount
      float re = 0.f, im = 0.f;
      if (n1 == 0) re = p;
      else if (n1 == 1) im = -p;
      else if (n1 == 2) re = -p;
      else im = p;
      sRe[j * 16 + m] = re;
      sIm[j * 16 + m] = im;
    }
  }
  __syncthreads();

  // ---- complex GEMM via V_WMMA_F32_16X16X4_F32 ----------------------------
  const int lane = tid & 31;
  const int wid  = tid >> 5;
  const int mA   = lane & 15;
  const int kOff = (lane >> 4) << 1;          // A/B frag: lanes 16-31 hold K+2
  const int mRowBase = (lane >> 4) << 3;      // C frag: lanes 16-31 hold M+8

#pragma unroll
  for (int t = 0; t < 2; ++t) {
    const int tile = wid + t * 8;
    const int nB = tile * 16 + (lane & 15);
    v8f acc_rr = {}, acc_ii = {}, acc_ri = {}, acc_ir = {};
    for (int k0 = 0; k0 < DIM; k0 += 4) {
      const int ka = k0 + kOff;
      v2f a_re, a_im, b_re, b_im;
      a_re.x = sRe[ka * 16 + mA];          a_re.y = sRe[(ka + 1) * 16 + mA];
      a_im.x = sIm[ka * 16 + mA];          a_im.y = sIm[(ka + 1) * 16 + mA];
      b_re.x = ut_re[ka * DIM + nB];       b_re.y = ut_re[(ka + 1) * DIM + nB];
      b_im.x = ut_im[ka * DIM + nB];       b_im.y = ut_im[(ka + 1) * DIM + nB];
      acc_rr = __builtin_amdgcn_wmma_f32_16x16x4_f32(false, a_re, false, b_re, (short)0, acc_rr, false, false);
      acc_ii = __builtin_amdgcn_wmma_f32_16x16x4_f32(false, a_im, false, b_im, (short)0, acc_ii, false, false);
      acc_ri = __builtin_amdgcn_wmma_f32_16x16x4_f32(false, a_re, false, b_im, (short)0, acc_ri, false, false);
      acc_ir = __builtin_amdgcn_wmma_f32_16x16x4_f32(false, a_im, false, b_re, (short)0, acc_ir, false, false);
    }
#pragma unroll
    for (int r = 0; r < 8; ++r) {
      float re = acc_rr[r] - acc_ii[r];
      float im = acc_ri[r] + acc_ir[r];
      sProb[(mRowBase + r) * PROBW + nB] = re * re + im * im;
    }
  }
  __syncthreads();

  // ---- expectation values: out[m][w] = sum_n probs[m][n] * (1-2*bit) ------
  if (tid < 128) {
    const int mm = tid >> 3, w = tid & 7;
    const int shift = 7 - w;
    float acc = 0.f;
    for (int n = 0; n < DIM; ++n) {
      float s = ((n >> shift) & 1) ? -1.f : 1.f;
      acc += s * sProb[mm * PROBW + n];
    }
    out[(base + mm) * 8 + w] = acc;
  }
}

// ============================================================================
extern "C" void kernel_launch(void* const* d_in, const int* in_sizes, int n_in,
                              void* d_out, int out_size, void* d_ws, size_t ws_size,
                              hipStream_t stream) {
  const float* x             = (const float*)d_in[0];   // [64,1024,8] f32
  const float* rx_params     = (const float*)d_in[1];   // [8]  f32
  const float* random_params = (const float*)d_in[2];   // [20] f32
  float* out = (float*)d_out;                           // [64,1024,8] f32

  float* ut_re = (float*)d_ws;                          // 256 KB
  float* ut_im = ut_re + DIM * DIM;                     // 256 KB  (512 KB total)

  const int nSamples = in_sizes[0] / N_WIRES;           // 65536

  build_unitary<<<DIM / COLS_PER_BLOCK, 256, 0, stream>>>(rx_params, random_params,
                                                          ut_re, ut_im);
  qsim_wmma<<<nSamples / 16, 256, 0, stream>>>(x, ut_re, ut_im, out);
}